// CNN_V1_32796370272431
// MI455X (gfx1250) — compile-verified
//
#include <hip/hip_runtime.h>

typedef __attribute__((ext_vector_type(2))) float v2f;
typedef __attribute__((ext_vector_type(4))) float v4f;
typedef __attribute__((ext_vector_type(8))) float v8f;

#define STEPN 4096
#define FEAT  64
#define HIDN  10

// ---------------------------------------------------------------------------
// Kernel A: synthesize per-channel smoothed kernels k2 (transposed to [s][f]).
// h  = relu(const_vec @ W1^T + b1)          [FEAT, HIDN]
// k1 = h @ W2^T + b2                        [FEAT, STEPN]   (kept in LDS)
// k2 = 5-tap smooth of k1 (band structure of `smooth`), stored k2T[s*64+f]
// ---------------------------------------------------------------------------
__global__ __launch_bounds__(256) void build_k2(
    const float* __restrict__ W1, const float* __restrict__ b1,
    const float* __restrict__ W2, const float* __restrict__ b2,
    const float* __restrict__ cvec, float* __restrict__ k2T) {
  const int f   = blockIdx.x;
  const int tid = threadIdx.x;
  __shared__ float k1[STEPN];

  const float c0 = cvec[0], c1 = cvec[1], c2 = cvec[2];
  float h[HIDN];
#pragma unroll
  for (int j = 0; j < HIDN; ++j) {
    float v = c0 * W1[f * HIDN * 3 + j * 3 + 0]
            + c1 * W1[f * HIDN * 3 + j * 3 + 1]
            + c2 * W1[f * HIDN * 3 + j * 3 + 2]
            + b1[f * HIDN + j];
    h[j] = v > 0.f ? v : 0.f;
  }

  const float* W2f = W2 + (size_t)f * STEPN * HIDN;
  const float* b2f = b2 + (size_t)f * STEPN;
  for (int s = tid; s < STEPN; s += 256) {
    float acc = b2f[s];
    const float* w = W2f + s * HIDN;
#pragma unroll
    for (int j = 0; j < HIDN; ++j) acc = fmaf(h[j], w[j], acc);
    k1[s] = acc;
  }
  __syncthreads();

  for (int s = tid; s < STEPN; s += 256) {
    float v = k1[s];
    if (s >= 1)         v = fmaf(0.8f, k1[s - 1], v);
    if (s >= 2)         v = fmaf(0.4f, k1[s - 2], v);
    if (s <= STEPN - 2) v = fmaf(0.8f, k1[s + 1], v);
    if (s <= STEPN - 3) v = fmaf(0.4f, k1[s + 2], v);
    k2T[s * FEAT + f] = v;
  }
}

// ---------------------------------------------------------------------------
// Kernel B: outs[b,f] = sum_s x[b,s,f] * k2[f,s].  HBM-bound stream of x.
// x[b] and k2T share the same flat [s*64+f] index -> per-lane scale is a
// simple same-index load (k2T = 1 MB, L2-resident).  float4 (b128) loads,
// non-temporal on x (read-once).  Feature quad per thread: f = 4*(tid&15).
// ---------------------------------------------------------------------------
__global__ __launch_bounds__(256) void stream_reduce(
    const v4f* __restrict__ x, const v4f* __restrict__ k2T,
    float* __restrict__ outs) {
  const int b   = blockIdx.x;
  const int tid = threadIdx.x;
  const v4f* xb = x + (size_t)b * (STEPN * FEAT / 4);

  v4f acc = (v4f)0.f;
#pragma unroll 4
  for (int i = tid; i < STEPN * FEAT / 4; i += 256) {
    v4f xv = __builtin_nontemporal_load(xb + i);
    v4f kv = k2T[i];
    acc.x = fmaf(xv.x, kv.x, acc.x);
    acc.y = fmaf(xv.y, kv.y, acc.y);
    acc.z = fmaf(xv.z, kv.z, acc.z);
    acc.w = fmaf(xv.w, kv.w, acc.w);
  }

  __shared__ v4f red[256];
  red[tid] = acc;
  __syncthreads();
  // threads sharing (tid & 15) hold the same feature quad; offsets are
  // multiples of 16 so the quad stays aligned through the tree.
  for (int off = 128; off >= 16; off >>= 1) {
    if (tid < off) {
      v4f a = red[tid], bv = red[tid + off];
      a.x += bv.x; a.y += bv.y; a.z += bv.z; a.w += bv.w;
      red[tid] = a;
    }
    __syncthreads();
  }
  if (tid < 16) {
    v4f r = red[tid];
    float* o = outs + (size_t)b * FEAT + 4 * tid;
    o[0] = r.x; o[1] = r.y; o[2] = r.z; o[3] = r.w;
  }
}

// ---------------------------------------------------------------------------
// Kernel C: fc head via v_wmma_f32_16x16x4_f32 (f32 precision preserved).
// One wave handles 16 batch rows: C[16x16] += A[16x4] * B[4x16], 16 k-steps
// over K=64.  B = fcW1^T zero-padded to N=16 via a 0/1 multiplicative mask
// (loads stay unconditional & in-bounds -> no EXEC-divergent load blocks).
// Epilogue: +fcb1, relu, dot with fcW2 across the 16 N-lanes (shfl_xor
// reduce), +fcb2.
// ISA layouts: A 16x4 f32 -> lane m=(lane&15), VGPR0/1 = K{0,1} (lanes 0-15)
// or K{2,3} (lanes 16-31).  D: VGPR j -> row j + 8*(lane>>4), col lane&15.
// ---------------------------------------------------------------------------
__global__ __launch_bounds__(32) void fc_head(
    const float* __restrict__ outs, const float* __restrict__ fcW1,
    const float* __restrict__ fcb1, const float* __restrict__ fcW2,
    const float* __restrict__ fcb2, float* __restrict__ out) {
  const int lane    = threadIdx.x;       // 0..31, wave32
  const int half    = lane >> 4;
  const int n       = lane & 15;         // output column (valid n < 10)
  const int m_a     = lane & 15;         // A-matrix row for this lane
  const int rowBase = blockIdx.x * 16;
  const int koff    = half * 2;

  const float mask = (n < HIDN) ? 1.0f : 0.0f;   // zero-pad columns 10..15
  const int   nc   = (n < HIDN) ? n : 0;         // clamped for in-bounds loads
  const float* arow = outs + (size_t)(rowBase + m_a) * FEAT;
  const float* bcol = fcW1 + (size_t)nc * FEAT;  // fcW1 is [10, 64]

  v8f c = {};
#pragma unroll
  for (int kk = 0; kk < 16; ++kk) {
    const int k0 = kk * 4 + koff;
    v2f a, b;
    a.x = arow[k0];
    a.y = arow[k0 + 1];
    b.x = bcol[k0]     * mask;   // unconditional b64 load + mask mul
    b.y = bcol[k0 + 1] * mask;
    c = __builtin_amdgcn_wmma_f32_16x16x4_f32(
        /*neg_a=*/false, a, /*neg_b=*/false, b,
        /*c_mod=*/(short)0, c, /*reuse_a=*/false, /*reuse_b=*/false);
  }

  const float w2    = fcW2[nc] * mask;
  const float bias1 = fcb1[nc];           // irrelevant for masked lanes
  const float bias2 = fcb2[0];
#pragma unroll
  for (int j = 0; j < 8; ++j) {
    float v = c[j] + bias1;
    v = v > 0.f ? v : 0.f;
    float t = v * w2;                     // masked lanes contribute 0
    t += __shfl_xor(t, 8);
    t += __shfl_xor(t, 4);
    t += __shfl_xor(t, 2);
    t += __shfl_xor(t, 1);
    if (n == 0) out[rowBase + half * 8 + j] = t + bias2;
  }
}

// ---------------------------------------------------------------------------
extern "C" void kernel_launch(void* const* d_in, const int* in_sizes, int n_in,
                              void* d_out, int out_size, void* d_ws, size_t ws_size,
                              hipStream_t stream) {
  const float* x    = (const float*)d_in[0];   // [256, 4096, 64]
  const float* W1   = (const float*)d_in[1];   // [64, 10, 3]
  const float* b1   = (const float*)d_in[2];   // [64, 10]
  const float* W2   = (const float*)d_in[3];   // [64, 4096, 10]
  const float* b2   = (const float*)d_in[4];   // [64, 4096]
  const float* fcW1 = (const float*)d_in[5];   // [10, 64]
  const float* fcb1 = (const float*)d_in[6];   // [10]
  const float* fcW2 = (const float*)d_in[7];   // [1, 10]
  const float* fcb2 = (const float*)d_in[8];   // [1]
  const float* cvec = (const float*)d_in[9];   // [3]
  // d_in[10] = smooth: banded 5-tap matrix; applied analytically in build_k2.

  float* k2T  = (float*)d_ws;                  // [4096 * 64] = 1 MB
  float* outs = k2T + STEPN * FEAT;            // [256 * 64]  = 64 KB
  float* out  = (float*)d_out;                 // [256]

  build_k2<<<dim3(FEAT), dim3(256), 0, stream>>>(W1, b1, W2, b2, cvec, k2T);
  stream_reduce<<<dim3(256), dim3(256), 0, stream>>>(
      (const v4f*)x, (const v4f*)k2T, outs);
  fc_head<<<dim3(16), dim3(32), 0, stream>>>(outs, fcW1, fcb1, fcW2, fcb2, out);
}